// UrbanSparse_13589276525128
// MI455X (gfx1250) — compile-verified
//
#include <hip/hip_runtime.h>
#include <hip/hip_bf16.h>

typedef __attribute__((ext_vector_type(2))) float v2f;
typedef __attribute__((ext_vector_type(8))) float v8f;

#define PAD_IDX 8192
#define H1D 512
#define H2D 64
#define BB  32
#define KK  32
#define LL  128

// ---------------------------------------------------------------------------
// Kernel 1: embedding_bag_mean for query (bags 0..31) and poi (bags 32..1055)
// One block per bag; indices staged in LDS; 2 dims per thread (256 thr * 2 = 512)
// ---------------------------------------------------------------------------
__global__ void __launch_bounds__(256) bag_kernel(const int* __restrict__ qbf,
                                                  const int* __restrict__ pbf,
                                                  const float* __restrict__ cb,
                                                  float* __restrict__ emb) {
  __shared__ int s_idx[LL];
  const int bag = blockIdx.x;
  const int tid = threadIdx.x;
  const int* ip = (bag < BB) ? (qbf + (size_t)bag * LL)
                             : (pbf + (size_t)(bag - BB) * LL);
  if (tid < LL) s_idx[tid] = ip[tid];
  __syncthreads();

  const int d0 = tid * 2;
  float s0 = 0.f, s1 = 0.f;
  int cnt = 0;
  for (int i = 0; i < LL; ++i) {
    int idx = s_idx[i];
    if (idx != PAD_IDX) {
      const float* row = cb + (size_t)idx * H1D + d0;
      s0 += row[0];
      s1 += row[1];
      ++cnt;
    }
  }
  float inv = (cnt > 0) ? (1.f / (float)cnt) : 0.f;
  float* orow = emb + (size_t)bag * H1D + d0;
  orow[0] = s0 * inv;
  orow[1] = s1 * inv;
}

// ---------------------------------------------------------------------------
// Kernel 2: hidden_layers  h = prelu(x,a0); h = prelu(h@W1+b1,a1); h = prelu(h@W2+b2,a2)
// 4 waves per block; each block computes a 16-row strip of the 64-wide output.
// Uses native fp32 WMMA: V_WMMA_F32_16X16X4_F32.
//   A 16x4 f32:  lane l holds m=l&15; v0 = K(+0 | +2 by l>>4), v1 = K(+1 | +3)
//   B 4x16 f32:  lane l holds n=l&15; same K split by lane half
//   C/D 16x16:   lane l holds n=l&15; VGPR r -> M = r + 8*(l>=16)
// ---------------------------------------------------------------------------
__global__ void __launch_bounds__(128) hidden_kernel(const float* __restrict__ x,
                                                     const float* __restrict__ W1,
                                                     const float* __restrict__ b1,
                                                     const float* __restrict__ W2,
                                                     const float* __restrict__ b2,
                                                     const float* __restrict__ a0p,
                                                     const float* __restrict__ a1p,
                                                     const float* __restrict__ a2p,
                                                     float* __restrict__ out) {
  __shared__ float h1[16][H2D + 2];   // +2 dword pad to spread LDS banks

  const int lane  = threadIdx.x & 31;
  const int wave  = threadIdx.x >> 5;          // 0..3 -> N tile
  const int mloc  = lane & 15;
  const int koff  = (lane >> 4) << 1;          // 0 or 2
  const int rbase = (lane >> 4) << 3;          // 0 or 8 (C/D row base)
  const int n     = wave * 16 + mloc;          // global column 0..63
  const int m0    = blockIdx.x * 16;

  const float a0 = a0p[0], a1 = a1p[0], a2 = a2p[0];

  // ---- GEMM 1: (16 x 512) @ (512 x 64), prelu(a0) fused on A-load ----
  const float* xr = x + (size_t)(m0 + mloc) * H1D;
  v8f acc = {0.f, 0.f, 0.f, 0.f, 0.f, 0.f, 0.f, 0.f};
  for (int k0 = 0; k0 < H1D; k0 += 4) {
    float ax = xr[k0 + koff];
    float ay = xr[k0 + koff + 1];
    ax = (ax >= 0.f) ? ax : a0 * ax;
    ay = (ay >= 0.f) ? ay : a0 * ay;
    v2f Av = {ax, ay};
    v2f Bv = {W1[(size_t)(k0 + koff) * H2D + n],
              W1[(size_t)(k0 + koff + 1) * H2D + n]};
    acc = __builtin_amdgcn_wmma_f32_16x16x4_f32(false, Av, false, Bv,
                                                (short)0, acc, false, false);
  }

  const float bias1 = b1[n];
#pragma unroll
  for (int r = 0; r < 8; ++r) {
    float v = acc[r] + bias1;
    v = (v >= 0.f) ? v : a1 * v;
    h1[rbase + r][n] = v;
  }
  __syncthreads();

  // ---- GEMM 2: (16 x 64) @ (64 x 64) from LDS ----
  v8f acc2 = {0.f, 0.f, 0.f, 0.f, 0.f, 0.f, 0.f, 0.f};
  for (int k0 = 0; k0 < H2D; k0 += 4) {
    v2f Av = {h1[mloc][k0 + koff], h1[mloc][k0 + koff + 1]};
    v2f Bv = {W2[(size_t)(k0 + koff) * H2D + n],
              W2[(size_t)(k0 + koff + 1) * H2D + n]};
    acc2 = __builtin_amdgcn_wmma_f32_16x16x4_f32(false, Av, false, Bv,
                                                 (short)0, acc2, false, false);
  }

  const float bias2 = b2[n];
#pragma unroll
  for (int r = 0; r < 8; ++r) {
    float v = acc2[r] + bias2;
    v = (v >= 0.f) ? v : a2 * v;
    out[(size_t)(m0 + rbase + r) * H2D + n] = v;
  }
}

// ---------------------------------------------------------------------------
// Kernel 3: intersection + eval dots + masked sum -> text_sim[b,k]
// One block per (b,k); thread l owns query slot l.
// ---------------------------------------------------------------------------
__global__ void __launch_bounds__(128) score_kernel(const int* __restrict__ qbf,
                                                    const int* __restrict__ pbf,
                                                    const float* __restrict__ qh,
                                                    const float* __restrict__ ph,
                                                    const float* __restrict__ q_eval,
                                                    const float* __restrict__ p_eval,
                                                    const float* __restrict__ ind_ap,
                                                    float* __restrict__ text_sim) {
  __shared__ int   s_p[LL];
  __shared__ float s_qh[H2D];
  __shared__ float s_ph[H2D];
  __shared__ float red[LL];

  const int bk  = blockIdx.x;     // b*32 + k
  const int b   = bk >> 5;
  const int tid = threadIdx.x;

  s_p[tid] = pbf[(size_t)bk * LL + tid];
  if (tid < H2D) {
    s_qh[tid] = qh[(size_t)b * H2D + tid];
    s_ph[tid] = ph[(size_t)bk * H2D + tid];
  }
  __syncthreads();

  const int qv = qbf[(size_t)b * LL + tid];
  bool hit = false;
  if (qv != PAD_IDX) {
#pragma unroll 8
    for (int i = 0; i < LL; ++i) hit |= (s_p[i] == qv);
  }

  float contrib = 0.f;
  if (hit) {
    const float* qc = q_eval + (size_t)qv * H2D;
    const float* pc = p_eval + (size_t)qv * H2D;
    float dq = 0.f, dp = 0.f;
#pragma unroll 8
    for (int h = 0; h < H2D; ++h) {
      dq = fmaf(s_qh[h], qc[h], dq);
      dp = fmaf(s_ph[h], pc[h], dp);
    }
    const float ia = ind_ap[0];
    dq = ((dq >= 0.f) ? dq : ia * dq) + 1.f;
    dp = ((dp >= 0.f) ? dp : ia * dp) + 1.f;
    contrib = dq * dp;
  }

  red[tid] = contrib;
  __syncthreads();
  for (int s = LL >> 1; s > 0; s >>= 1) {
    if (tid < s) red[tid] += red[tid + s];
    __syncthreads();
  }
  if (tid == 0) text_sim[bk] = red[0];
}

// ---------------------------------------------------------------------------
// Kernel 4: per-row max normalization + distance + sigmoid epilogue
// One wave (32 lanes) per query row b; wave32 shuffle max.
// ---------------------------------------------------------------------------
__global__ void __launch_bounds__(32) final_kernel(const float* __restrict__ text_sim,
                                                   const float* __restrict__ qloc,
                                                   const float* __restrict__ ploc,
                                                   const float* __restrict__ ap,
                                                   const float* __restrict__ bp,
                                                   const float* __restrict__ cp,
                                                   const float* __restrict__ dp,
                                                   float* __restrict__ out) {
  const int b = blockIdx.x;
  const int k = threadIdx.x;

  float ts = text_sim[(size_t)b * KK + k];
  float mx = ts;
#pragma unroll
  for (int off = 16; off > 0; off >>= 1) {
    float o = __shfl_xor(mx, off, 32);
    mx = fmaxf(mx, o);
  }
  float tsn = (2.f * ts - mx) / (mx + 1e-6f);

  float dx = qloc[b * 2 + 0] - ploc[((size_t)b * KK + k) * 2 + 0];
  float dy = qloc[b * 2 + 1] - ploc[((size_t)b * KK + k) * 2 + 1];
  float dist = sqrtf(dx * dx + dy * dy);
  float dist_sim = -logf(dist + 1.f);

  const float A = ap[0], Bc = bp[0], Cc = cp[0], Dc = dp[0];
  float sig = 1.f / (1.f + expf(-(A * tsn + Bc)));
  out[(size_t)b * KK + k] = (Cc - sig) * (dist_sim - Dc);
}

// ---------------------------------------------------------------------------
// Host-side launch
// ---------------------------------------------------------------------------
extern "C" void kernel_launch(void* const* d_in, const int* in_sizes, int n_in,
                              void* d_out, int out_size, void* d_ws, size_t ws_size,
                              hipStream_t stream) {
  const int*   qbf  = (const int*)  d_in[0];   // (32,128)
  const int*   pbf  = (const int*)  d_in[1];   // (32,32,128)
  const float* qloc = (const float*)d_in[2];   // (32,2)
  const float* ploc = (const float*)d_in[3];   // (32,32,2)
  const float* cb   = (const float*)d_in[4];   // (8193,512)
  const float* qW1  = (const float*)d_in[5];
  const float* qb1  = (const float*)d_in[6];
  const float* qW2  = (const float*)d_in[7];
  const float* qb2  = (const float*)d_in[8];
  const float* pW1  = (const float*)d_in[9];
  const float* pb1  = (const float*)d_in[10];
  const float* pW2  = (const float*)d_in[11];
  const float* pb2  = (const float*)d_in[12];
  const float* qev  = (const float*)d_in[13];  // (8193,64)
  const float* pev  = (const float*)d_in[14];  // (8193,64)
  const float* qa0  = (const float*)d_in[15];
  const float* qa1  = (const float*)d_in[16];
  const float* qa2  = (const float*)d_in[17];
  const float* pa0  = (const float*)d_in[18];
  const float* pa1  = (const float*)d_in[19];
  const float* pa2  = (const float*)d_in[20];
  const float* inda = (const float*)d_in[21];
  const float* sa   = (const float*)d_in[22];
  const float* sb   = (const float*)d_in[23];
  const float* sc   = (const float*)d_in[24];
  const float* sd   = (const float*)d_in[25];

  // workspace layout (floats)
  float* ws   = (float*)d_ws;
  float* emb  = ws;                                  // 1056*512 (query rows 0..31, poi rows 32..1055)
  float* qhid = emb  + (size_t)(BB + BB * KK) * H1D; // 32*64
  float* phid = qhid + (size_t)BB * H2D;             // 1024*64
  float* tsim = phid + (size_t)BB * KK * H2D;        // 1024

  // 1) embedding bags (32 query + 1024 poi)
  bag_kernel<<<BB + BB * KK, 256, 0, stream>>>(qbf, pbf, cb, emb);

  // 2) MLPs (fp32 WMMA): query (32 rows -> 2 blocks), poi (1024 rows -> 64 blocks)
  hidden_kernel<<<BB / 16, 128, 0, stream>>>(emb, qW1, qb1, qW2, qb2,
                                             qa0, qa1, qa2, qhid);
  hidden_kernel<<<(BB * KK) / 16, 128, 0, stream>>>(emb + (size_t)BB * H1D,
                                                    pW1, pb1, pW2, pb2,
                                                    pa0, pa1, pa2, phid);

  // 3) intersection scoring -> text_sim
  score_kernel<<<BB * KK, LL, 0, stream>>>(qbf, pbf, qhid, phid, qev, pev,
                                           inda, tsim);

  // 4) normalization + distance epilogue -> out (32x32)
  final_kernel<<<BB, KK, 0, stream>>>(tsim, qloc, ploc, sa, sb, sc, sd,
                                      (float*)d_out);
}